// BaseModule_35287451304527
// MI455X (gfx1250) — compile-verified
//
#include <hip/hip_runtime.h>
#include <math.h>

// ---------------- problem constants (match reference) ----------------
#define Hdim 256
#define Wdim 256
#define Npix (Hdim * Wdim)   // 65536
#define Bdim 2
#define Vdim 3
#define Cdim 32
#define BV   (Bdim * Vdim)   // 6
#define RADIUS 1.5f
#define TAU    1.0f
#define EPSF   1e-8f

typedef float v2f __attribute__((ext_vector_type(2)));
typedef float v8f __attribute__((ext_vector_type(8)));

// ---------------------------------------------------------------------
// 4x4 inverse (Gauss-Jordan with partial pivoting) — runs on one lane,
// only for the 2 K matrices; negligible cost.
// ---------------------------------------------------------------------
__device__ void inv4x4(const float* in, float* out) {
    float a[4][8];
    for (int i = 0; i < 4; ++i)
        for (int j = 0; j < 4; ++j) {
            a[i][j]     = in[i * 4 + j];
            a[i][j + 4] = (i == j) ? 1.0f : 0.0f;
        }
    for (int col = 0; col < 4; ++col) {
        int piv = col; float best = fabsf(a[col][col]);
        for (int r = col + 1; r < 4; ++r) {
            float v = fabsf(a[r][col]);
            if (v > best) { best = v; piv = r; }
        }
        if (piv != col)
            for (int j = 0; j < 8; ++j) { float t = a[col][j]; a[col][j] = a[piv][j]; a[piv][j] = t; }
        float inv = 1.0f / a[col][col];
        for (int j = 0; j < 8; ++j) a[col][j] *= inv;
        for (int r = 0; r < 4; ++r) if (r != col) {
            float f = a[r][col];
            for (int j = 0; j < 8; ++j) a[r][j] -= f * a[col][j];
        }
    }
    for (int i = 0; i < 4; ++i)
        for (int j = 0; j < 4; ++j) out[i * 4 + j] = a[i][j + 4];
}

// ---------------------------------------------------------------------
// 4x4 @ 4x4 product via V_WMMA_F32_16X16X4_F32 (one full wave32).
// A (16x4, 32-bit layout): lane L<16 holds row L, vgpr0=K0 vgpr1=K1;
//                          lane L+16 holds row L, vgpr0=K2 vgpr1=K3.
// B (4x16): lane j<16 holds col j rows {0,1}; lane j+16 rows {2,3}.
// D (16x16): lane j holds col j; vgpr r = row r (lanes<16) / r+8.
// Product lands in rows 0..3 (D[0..3]) at lanes 0..3. Padding rows/cols
// are zero so the extra MACs are exact no-ops.
// ---------------------------------------------------------------------
__device__ __forceinline__ void mm4_wmma(const float* L, const float* R,
                                         float* out, int lane) {
    const int idx  = lane & 15;
    const int half = lane >> 4;
#if __has_builtin(__builtin_amdgcn_wmma_f32_16x16x4_f32)
    v2f A, Bm;
    A.x  = (idx < 4) ? L[idx * 4 + 2 * half + 0] : 0.0f;
    A.y  = (idx < 4) ? L[idx * 4 + 2 * half + 1] : 0.0f;
    Bm.x = (idx < 4) ? R[(2 * half + 0) * 4 + idx] : 0.0f;
    Bm.y = (idx < 4) ? R[(2 * half + 1) * 4 + idx] : 0.0f;
    v8f Cz = {0.f, 0.f, 0.f, 0.f, 0.f, 0.f, 0.f, 0.f};
    v8f D = __builtin_amdgcn_wmma_f32_16x16x4_f32(
        false, A, false, Bm, (short)0, Cz, false, false);
    __syncthreads();
    if (lane < 4) {            // lane = column j, D[i] = row i
        out[0 * 4 + lane] = D[0];
        out[1 * 4 + lane] = D[1];
        out[2 * 4 + lane] = D[2];
        out[3 * 4 + lane] = D[3];
    }
#else
    __syncthreads();
    if (lane < 16) {
        int i = lane >> 2, j = lane & 3;
        float s = 0.f;
        for (int k = 0; k < 4; ++k) s += L[i * 4 + k] * R[k * 4 + j];
        out[i * 4 + j] = s;
    }
#endif
    __syncthreads();
}

// ---------------------------------------------------------------------
// Kernel 1: fuse M[bv] = K[b] @ dstRT[b] @ srcRTinv[b,v] @ inv(K[b]).
// One wave32 per (b,v); three chained f32 WMMAs.
// ---------------------------------------------------------------------
__global__ __launch_bounds__(32)
void fuse_mats_kernel(const float* __restrict__ K,
                      const float* __restrict__ srcRTinv,
                      const float* __restrict__ dstRT,
                      float* __restrict__ Mout) {
    const int bv = blockIdx.x;
    const int b  = bv / Vdim;
    const int lane = threadIdx.x;
    __shared__ float sK[16], sKinv[16], sDst[16], sSrc[16];
    __shared__ float P1[16], P2[16], Mf[16];
    if (lane == 0) {
        for (int i = 0; i < 16; ++i) {
            sK[i]   = K[b * 16 + i];
            sDst[i] = dstRT[b * 16 + i];    // dst_RTs[:,0]
            sSrc[i] = srcRTinv[bv * 16 + i];
        }
        inv4x4(sK, sKinv);
    }
    __syncthreads();
    mm4_wmma(sSrc, sKinv, P1, lane);   // srcRTinv @ K^-1
    mm4_wmma(sDst, P1,    P2, lane);   // dstRT @ (..)
    mm4_wmma(sK,   P2,    Mf, lane);   // K @ (..)
    if (lane < 16) Mout[bv * 16 + lane] = Mf[lane];
}

// ---------------------------------------------------------------------
// Kernel 2: zero an accumulator region.
// ---------------------------------------------------------------------
__global__ __launch_bounds__(256)
void zero_kernel(float* __restrict__ p, long n) {
    long i = (long)blockIdx.x * blockDim.x + threadIdx.x;
    if (i < n) p[i] = 0.0f;
}

// ---------------------------------------------------------------------
// Kernel 3: projective splat-scatter. One thread per source pixel.
// Accumulators live in L2 (53 MB << 192 MB): no-return f32 atomics.
// ---------------------------------------------------------------------
__global__ __launch_bounds__(256)
void splat_kernel(const float* __restrict__ depths,
                  const float* __restrict__ feats,
                  const float* __restrict__ Mws,
                  float* __restrict__ wsum,
                  float* __restrict__ facc,
                  float* __restrict__ zacc) {
    const int bv = blockIdx.y;
    const int p  = blockIdx.x * 256 + threadIdx.x;

    __shared__ float sM[16];
    if (threadIdx.x < 16) sM[threadIdx.x] = Mws[bv * 16 + threadIdx.x];
    __syncthreads();

    // Warm the cachelines of every 4th feature channel, 8 of 32 (global_prefetch_b8), while the
    // transform ALU work runs.
    const long fbase = ((long)bv * Cdim) * Npix + p;
#pragma unroll
    for (int c = 0; c < Cdim; c += 4)
        __builtin_prefetch(&feats[fbase + (long)c * Npix], 0, 3);

    const int row = p >> 8;
    const int col = p & 255;
    const float d = depths[(long)bv * Npix + p];

    const float x = (col / 255.0f) * 2.0f - 1.0f;
    const float y = (row / 255.0f) * 2.0f - 1.0f;
    const float p0 = x * d, p1 = y * d, p2 = d;

    const float prx = sM[0] * p0 + sM[1] * p1 + sM[2]  * p2 + sM[3];
    const float pry = sM[4] * p0 + sM[5] * p1 + sM[6]  * p2 + sM[7];
    const float prz = sM[8] * p0 + sM[9] * p1 + sM[10] * p2 + sM[11];

    if (prz <= 0.0f) return;   // reference: valid requires z > 0

    const float sx = prx / (prz + EPSF);
    const float sy = pry / (prz + EPSF);
    const float px = (sx + 1.0f) * 0.5f * (Wdim - 1);
    const float py = (sy + 1.0f) * 0.5f * (Hdim - 1);
    const float rpx = rintf(px);        // jnp.round = half-to-even
    const float rpy = rintf(py);
    const float wd  = expf(-TAU * prz);

    float f[Cdim];
#pragma unroll
    for (int c = 0; c < Cdim; ++c) f[c] = feats[fbase + (long)c * Npix];

#pragma unroll
    for (int dy = -1; dy <= 1; ++dy) {
#pragma unroll
        for (int dx = -1; dx <= 1; ++dx) {
            const float fx = rpx + (float)dx;
            const float fy = rpy + (float)dy;
            const float ddx = fx - px;
            const float ddy = fy - py;
            const float dist = sqrtf(ddx * ddx + ddy * ddy + EPSF);
            float wr = 1.0f - dist / RADIUS;
            wr = fmaxf(wr, 0.0f);
            const float w = wr * wd;
            const bool valid = (fx >= 0.0f) && (fx < (float)Wdim) &&
                               (fy >= 0.0f) && (fy < (float)Hdim);
            if (valid && w > 0.0f) {
                const int q = (int)fy * Wdim + (int)fx;
                atomicAdd(&wsum[(long)bv * Npix + q], w);
                atomicAdd(&zacc[(long)bv * Npix + q], prz * w);
                const long ob = ((long)bv * Cdim) * Npix + q;
#pragma unroll
                for (int c = 0; c < Cdim; ++c)
                    atomicAdd(&facc[ob + (long)c * Npix], f[c] * w);
            }
        }
    }
}

// ---------------------------------------------------------------------
// Kernel 4: normalize by weight sum (one thread per destination pixel).
// ---------------------------------------------------------------------
__global__ __launch_bounds__(256)
void normalize_kernel(const float* __restrict__ wsum,
                      float* __restrict__ facc,
                      float* __restrict__ zacc) {
    const long i = (long)blockIdx.x * 256 + threadIdx.x;   // over BV*Npix
    const int bv = (int)(i >> 16);
    const int q  = (int)(i & (Npix - 1));
    const float inv = 1.0f / (wsum[i] + EPSF);
    const long ob = ((long)bv * Cdim) * Npix + q;
#pragma unroll
    for (int c = 0; c < Cdim; ++c) facc[ob + (long)c * Npix] *= inv;
    zacc[i] *= inv;
}

// ---------------------------------------------------------------------
// Launch. Inputs (reference order): depths, feats, K, src_RTs,
// src_RTinvs, dst_RTs, dst_RTinvs.  Output: prj_f (B,V,C,H,W) then
// prj_d (B,V,1,H,W), both float32, accumulated in-place in d_out.
// Workspace: 96 floats fused matrices + BV*Npix floats weight sums.
// ---------------------------------------------------------------------
extern "C" void kernel_launch(void* const* d_in, const int* in_sizes, int n_in,
                              void* d_out, int out_size, void* d_ws, size_t ws_size,
                              hipStream_t stream) {
    const float* depths   = (const float*)d_in[0];
    const float* feats    = (const float*)d_in[1];
    const float* K        = (const float*)d_in[2];
    const float* srcRTinv = (const float*)d_in[4];
    const float* dstRT    = (const float*)d_in[5];

    float* facc = (float*)d_out;                       // BV*C*N feats accum
    float* zacc = facc + (long)BV * Cdim * Npix;       // BV*N depth accum
    float* Mws  = (float*)d_ws;                        // 96 floats
    float* wsum = Mws + 96;                            // BV*N weight accum

    const long n_out = (long)BV * Cdim * Npix + (long)BV * Npix;
    zero_kernel<<<(int)((n_out + 255) / 256), 256, 0, stream>>>(facc, n_out);
    zero_kernel<<<(BV * Npix + 255) / 256, 256, 0, stream>>>(wsum, (long)BV * Npix);

    fuse_mats_kernel<<<BV, 32, 0, stream>>>(K, srcRTinv, dstRT, Mws);

    dim3 sgrid(Npix / 256, BV);
    splat_kernel<<<sgrid, 256, 0, stream>>>(depths, feats, Mws, wsum, facc, zacc);

    normalize_kernel<<<(BV * Npix) / 256, 256, 0, stream>>>(wsum, facc, zacc);
}